// SparseToDense_82437602279977
// MI455X (gfx1250) — compile-verified
//
#include <hip/hip_runtime.h>
#include <stdint.h>

// SparseToDense scatter: features [400000,32] f32 + flat_idx [400000] int
// -> dense [4,32,64,64,64] f32 (NCDHW), duplicate indices accumulate (atomic add).
//
// MI455X strategy: 128MB output is resident in the 192MB L2, so the 12.8M
// scattered 4B atomic adds resolve at L2 atomic units; HBM sees only clean
// streams (zero-fill write, feature read, final writeback). Memory-bound,
// ~13us ideal at 23.3 TB/s.

#define SD_C        32
#define SD_DHW      (64 * 64 * 64)        // 262144 = 1<<18
#define SD_NACTIVE  400000
#define SD_GROUP    4                     // sites per wave iteration
#define SD_NGROUPS  (SD_NACTIVE / SD_GROUP)   // 100000 exactly
#define SD_OUT_F32  (4 * SD_C * SD_DHW)   // 33554432 floats = 128 MB

// ---------------------------------------------------------------------------
// Zero-fill the dense output with wide stores (lines land dirty in L2, ready
// to absorb the scatter atomics).
__global__ __launch_bounds__(256) void sd_zero_kernel(float4* __restrict__ out, int n4) {
    int i = blockIdx.x * blockDim.x + threadIdx.x;
    const int stride = gridDim.x * blockDim.x;
    const float4 z = make_float4(0.f, 0.f, 0.f, 0.f);
    for (; i < n4; i += stride) out[i] = z;
}

// ---------------------------------------------------------------------------
// gfx1250 async copy: each lane moves 16B global -> LDS; tracked by ASYNCcnt.
__device__ __forceinline__ void sd_async_b128(uint32_t lds_byte_addr,
                                              uint32_t gbl_byte_off,
                                              const float* gbase) {
    asm volatile("global_load_async_to_lds_b128 %0, %1, %2"
                 :: "v"(lds_byte_addr), "v"(gbl_byte_off),
                    "s"((unsigned long long)(uintptr_t)gbase)
                 : "memory");
}

// Non-returning f32 global atomic add (TH[0]=0 -> no return, STOREcnt path).
__device__ __forceinline__ void sd_atomic_add_f32(float* base, uint32_t byte_off, float v) {
    asm volatile("global_atomic_add_f32 %0, %1, %2"
                 :: "v"(byte_off), "v"(v),
                    "s"((unsigned long long)(uintptr_t)base)
                 : "memory");
}

// ---------------------------------------------------------------------------
// One wave32 per group of 4 sites; lane == channel. Feature rows are staged
// through LDS with ping-pong double-buffered async copies so the next group's
// 512B block streams in while the current group's 128 atomics issue.
__global__ __launch_bounds__(256) void sd_scatter_kernel(const float* __restrict__ feats,
                                                         const int*   __restrict__ idx,
                                                         float*       __restrict__ out) {
    // 8 waves/block x 2 buffers x 128 floats (512B) = 8KB LDS
    __shared__ alignas(16) float smem[8 * 2 * 128];

    const int wave = threadIdx.x >> 5;
    const int lane = threadIdx.x & 31;

    // Low 32 bits of the flat pointer to LDS == wave-relative LDS byte offset.
    const uint32_t lds_wave_base =
        (uint32_t)(uintptr_t)(void*)smem + (uint32_t)wave * 1024u + (uint32_t)lane * 16u;

    const int nwaves = (int)gridDim.x * 8;
    // g is wave-uniform by construction; pin it to SGPRs so idx loads scalarize.
    int g = __builtin_amdgcn_readfirstlane((int)blockIdx.x * 8 + wave);
    if (g >= SD_NGROUPS) return;

    uint32_t buf = 0;
    sd_async_b128(lds_wave_base, (uint32_t)g * 512u, feats);   // prime buffer 0

    while (true) {
        const int  gn   = g + nwaves;
        const bool more = gn < SD_NGROUPS;
        if (more) {
            sd_async_b128(lds_wave_base + ((buf ^ 1u) * 512u), (uint32_t)gn * 512u, feats);
            asm volatile("s_wait_asynccnt 1" ::: "memory");    // current buffer ready
        } else {
            asm volatile("s_wait_asynccnt 0" ::: "memory");
        }

        const float* mybuf = &smem[wave * 256 + (int)buf * 128];
        const int site0 = g * SD_GROUP;
#pragma unroll
        for (int s = 0; s < SD_GROUP; ++s) {
            const uint32_t f = (uint32_t)idx[site0 + s];       // uniform -> s_load
            // out elem = b*(C*DHW) + c*DHW + spatial ; DHW=2^18, C*DHW=2^23
            const uint32_t oelem = ((f >> 18) << 23) | ((uint32_t)lane << 18) | (f & 0x3FFFFu);
            const float v = mybuf[s * 32 + lane];
            sd_atomic_add_f32(out, oelem * 4u, v);
        }

        if (!more) break;
        buf ^= 1u;
        g = gn;
    }
}

// ---------------------------------------------------------------------------
extern "C" void kernel_launch(void* const* d_in, const int* in_sizes, int n_in,
                              void* d_out, int out_size, void* d_ws, size_t ws_size,
                              hipStream_t stream) {
    const float* feats = (const float*)d_in[0];   // [400000, 32] f32
    const int*   idx   = (const int*)d_in[1];     // [400000] int (values < 2^20)
    float*       out   = (float*)d_out;           // [4,32,64,64,64] f32

    // 1) zero the 128MB dense grid (8.39M float4 stores)
    sd_zero_kernel<<<4096, 256, 0, stream>>>((float4*)out, SD_OUT_F32 / 4);

    // 2) scatter-add: 1024 blocks x 8 waves = 8192 waves, ~12 groups each
    sd_scatter_kernel<<<1024, 256, 0, stream>>>(feats, idx, out);
}